// KMeansClustering_34471407517798
// MI455X (gfx1250) — compile-verified
//
#include <hip/hip_runtime.h>
#include <hip/hip_bf16.h>
#include <float.h>

typedef __attribute__((ext_vector_type(2))) float v2f;
typedef __attribute__((ext_vector_type(8))) float v8f;

#define N_PTS   30000
#define DIM     64
#define K_CL    500
#define K_PAD   512
#define N_ITERS 1000
#define EPSV    1e-6f

// ---------------- init: cents <- init_centroids (padded), c_sq, zero sums/counts
__global__ void km_init(const float* __restrict__ init_c,
                        float* __restrict__ cents, float* __restrict__ c_sq,
                        float* __restrict__ sums, float* __restrict__ counts) {
    int k = blockIdx.x;      // 0..511
    int d = threadIdx.x;     // 0..63
    float v = 0.0f;
    if (k < K_CL) v = init_c[k * DIM + d];
    cents[k * DIM + d] = v;
    sums[k * DIM + d]  = 0.0f;
    __shared__ float red[DIM];
    red[d] = v * v;
    __syncthreads();
    if (d == 0) {
        float s = 0.0f;
        for (int i = 0; i < DIM; ++i) s += red[i];
        c_sq[k]   = (k < K_CL) ? s : FLT_MAX;   // pad rows can never win argmin
        counts[k] = 0.0f;
    }
}

// ---------------- assign: centroids staged in LDS (128KB of the 320KB WGP LDS),
// per wave TWO 16-point tiles sharing B fragments; B preloaded per K-chunk so
// 32 WMMAs issue per wait group.
__global__ __launch_bounds__(256, 1) void
km_assign(const float* __restrict__ embeds, const float* __restrict__ cents,
          const float* __restrict__ c_sq, int* __restrict__ idx_int,
          float* __restrict__ idx_out) {
    __shared__ float lds_c[K_PAD * DIM];   // 128 KB
    __shared__ float lds_q[K_PAD];         // 2 KB (c_sq)

    // cooperative copy: 8192 float4 over 256 threads
    {
        const float4* src = (const float4*)cents;
        float4*       dst = (float4*)lds_c;
#pragma unroll
        for (int i = 0; i < (K_PAD * DIM / 4) / 256; ++i) {
            int idx = threadIdx.x + i * 256;
            dst[idx] = src[idx];
        }
        lds_q[threadIdx.x]       = c_sq[threadIdx.x];
        lds_q[threadIdx.x + 256] = c_sq[threadIdx.x + 256];
    }
    __syncthreads();

    const int wave = threadIdx.x >> 5;
    const int lane = threadIdx.x & 31;
    const int pair = blockIdx.x * 8 + wave;       // 32-point super-tile
    if (pair * 32 >= N_PTS) return;               // wave-uniform; no barriers follow

    const int m0   = pair * 32;
    const int half = lane >> 4;                   // 0: K-subdims {0,1}, 1: {2,3}
    const int lrow = lane & 15;
    const int dsel = half * 2;

    // A fragments for tile0 (rows m0..m0+15) and tile1 (rows m0+16..m0+31)
    int row0 = m0 + lrow;
    int row1 = m0 + 16 + lrow;
    if (row1 >= N_PTS) row1 = 0;                  // tail: safe reads, stores guarded later
    const float* arow0 = embeds + (size_t)row0 * DIM + dsel;
    const float* arow1 = embeds + (size_t)row1 * DIM + dsel;
    v2f a0[16], a1[16];
#pragma unroll
    for (int kk = 0; kk < 16; ++kk) {
        float2 t0 = *(const float2*)(arow0 + kk * 4);
        float2 t1 = *(const float2*)(arow1 + kk * 4);
        v2f v0; v0.x = t0.x; v0.y = t0.y; a0[kk] = v0;
        v2f v1; v1.x = t1.x; v1.y = t1.y; a1[kk] = v1;
    }

    float minv0[8], minv1[8];
    int   mini0[8], mini1[8];
#pragma unroll
    for (int r = 0; r < 8; ++r) {
        minv0[r] = FLT_MAX; mini0[r] = 0;
        minv1[r] = FLT_MAX; mini1[r] = 0;
    }

    const float* bbase = lds_c + lrow * DIM + dsel;

    for (int chunk = 0; chunk < K_PAD / 16; ++chunk) {
        const int c0 = chunk * 16;
        const float* brow = bbase + c0 * DIM;

        // preload all 16 B fragments for this chunk (ds_load_b64 burst)
        v2f b[16];
#pragma unroll
        for (int kk = 0; kk < 16; ++kk) {
            float2 t = *(const float2*)(brow + kk * 4);
            v2f bv; bv.x = t.x; bv.y = t.y;
            b[kk] = bv;
        }

        v8f acc0 = {};
        v8f acc1 = {};
#pragma unroll
        for (int kk = 0; kk < 16; ++kk) {
            acc0 = __builtin_amdgcn_wmma_f32_16x16x4_f32(
                false, a0[kk], false, b[kk], (short)0, acc0, false, false);
            acc1 = __builtin_amdgcn_wmma_f32_16x16x4_f32(
                false, a1[kk], false, b[kk], (short)0, acc1, false, false);
        }

        // lane j holds centroid c0+(j&15); VGPR r -> point row r + half*8
        const int   cidx = c0 + lrow;
        const float csq  = lds_q[cidx];
#pragma unroll
        for (int r = 0; r < 8; ++r) {
            float s0 = fmaf(-2.0f, acc0[r], csq);   // ||e||^2 dropped: argmin-invariant
            float s1 = fmaf(-2.0f, acc1[r], csq);
            if (s0 < minv0[r]) { minv0[r] = s0; mini0[r] = cidx; }
            if (s1 < minv1[r]) { minv1[r] = s1; mini1[r] = cidx; }
        }
    }

    // reduce (min, argmin) across the 16 lanes of each half; tile0 then tile1
#pragma unroll
    for (int r = 0; r < 8; ++r) {
        float v = minv0[r];
        int   i = mini0[r];
#pragma unroll
        for (int off = 1; off < 16; off <<= 1) {
            float vo = __shfl_xor(v, off, 32);
            int   io = __shfl_xor(i, off, 32);
            if (vo < v || (vo == v && io < i)) { v = vo; i = io; }
        }
        if (lrow == 0) {
            int m = m0 + r + half * 8;
            idx_int[m] = i;
            idx_out[m] = (float)i;
        }
    }
#pragma unroll
    for (int r = 0; r < 8; ++r) {
        float v = minv1[r];
        int   i = mini1[r];
#pragma unroll
        for (int off = 1; off < 16; off <<= 1) {
            float vo = __shfl_xor(v, off, 32);
            int   io = __shfl_xor(i, off, 32);
            if (vo < v || (vo == v && io < i)) { v = vo; i = io; }
        }
        if (lrow == 0) {
            int m = m0 + 16 + r + half * 8;
            if (m < N_PTS) {
                idx_int[m] = i;
                idx_out[m] = (float)i;
            }
        }
    }
}

// ---------------- accumulate: segment-sum via L2 atomics (128KB hot region)
__global__ __launch_bounds__(256) void
km_accum(const float* __restrict__ embeds, const int* __restrict__ idx_int,
         float* __restrict__ sums, float* __restrict__ counts) {
    int t = blockIdx.x * 256 + threadIdx.x;
    int p = t >> 6;
    int d = t & 63;
    if (p >= N_PTS) return;
    int c = idx_int[p];
    atomicAdd(&sums[(size_t)c * DIM + d], embeds[(size_t)p * DIM + d]);
    if (d == 0) atomicAdd(&counts[c], 1.0f);
}

// ---------------- update: new centroid, c_sq, stream outputs, re-zero accumulators
__global__ void km_update(float* __restrict__ cents, float* __restrict__ c_sq,
                          float* __restrict__ sums, float* __restrict__ counts,
                          float* __restrict__ out_cents, float* __restrict__ out_counts) {
    int k = blockIdx.x;      // 0..499
    int d = threadIdx.x;     // 0..63
    float cnt = counts[k];
    float nc  = sums[k * DIM + d] / (cnt + EPSV);
    cents[k * DIM + d]     = nc;
    out_cents[k * DIM + d] = nc;
    sums[k * DIM + d]      = 0.0f;
    __shared__ float red[DIM];
    red[d] = nc * nc;
    __syncthreads();
    if (d == 0) {
        float s = 0.0f;
        for (int i = 0; i < DIM; ++i) s += red[i];
        c_sq[k]       = s;
        out_counts[k] = cnt;   // last iteration's value survives in d_out
        counts[k]     = 0.0f;
    }
}

extern "C" void kernel_launch(void* const* d_in, const int* in_sizes, int n_in,
                              void* d_out, int out_size, void* d_ws, size_t ws_size,
                              hipStream_t stream) {
    const float* embeds = (const float*)d_in[0];   // [30000,64] fp32
    const float* init_c = (const float*)d_in[1];   // [500,64]   fp32

    float* out        = (float*)d_out;
    float* out_cents  = out;                        // 500*64 = 32000
    float* out_idx    = out + 32000;                // 30000 (indices as float)
    float* out_counts = out + 62000;                // 500

    float* ws      = (float*)d_ws;                  // ~383 KB used
    float* cents   = ws;                            // 512*64
    float* c_sq    = ws + K_PAD * DIM;              // 512
    float* sums    = c_sq + K_PAD;                  // 512*64
    float* counts  = sums + K_PAD * DIM;            // 512
    int*   idx_int = (int*)(counts + K_PAD);        // 30000

    km_init<<<K_PAD, DIM, 0, stream>>>(init_c, cents, c_sq, sums, counts);

    const int n_pairs = (N_PTS + 31) / 32;          // 938 super-tiles (32 points each)
    const int a_grid  = (n_pairs + 7) / 8;          // 8 waves/block, 1 super-tile/wave

    for (int it = 0; it < N_ITERS; ++it) {
        km_assign<<<a_grid, 256, 0, stream>>>(embeds, cents, c_sq, idx_int, out_idx);
        km_accum<<<N_PTS / 4, 256, 0, stream>>>(embeds, idx_int, sums, counts);
        km_update<<<K_CL, DIM, 0, stream>>>(cents, c_sq, sums, counts,
                                            out_cents, out_counts);
    }
}